// BranchRoute_55241869361851
// MI455X (gfx1250) — compile-verified
//
#include <hip/hip_runtime.h>
#include <hip/hip_bf16.h>

typedef __attribute__((ext_vector_type(2))) float v2f;
typedef __attribute__((ext_vector_type(4))) float v4f;
typedef __attribute__((ext_vector_type(8))) float v8f;

#define DIM   4096
#define NROWS 8192

// ---------------------------------------------------------------------------
// Pass 1: gate logits via V_WMMA_F32_16X16X4_F32.
// One wave (32 lanes) computes a 16-row tile of score = x @ W.
// A (16x4 f32): lane L -> row m = L&15, kh = L>>4; VGPR pair holds columns
//   (k + 2*kh, k + 2*kh + 1)  => contiguous float2 load from x.
// B (4x16 f32): lane L -> K row 2*kh(+1), N col = m; only N<2 carry W, rest 0.
// C/D (16x16 f32): VGPR r, lane L -> element (M = r + 8*kh, N = L&15).
// ---------------------------------------------------------------------------
__global__ __launch_bounds__(256) void gate_wmma_kernel(
    const float* __restrict__ x,
    const float* __restrict__ gate_w,   // [DIM][2] row-major
    const float* __restrict__ gate_b,   // [2]
    float* __restrict__ masks)          // [2][NROWS] of {0,1}
{
    __shared__ float w_lds[DIM * 2];    // 32 KB

    const int tid = threadIdx.x;
    // Cooperative vectorized stage of gate_w into LDS (8192 floats).
    for (int i = tid * 4; i < DIM * 2; i += 256 * 4) {
        *(v4f*)&w_lds[i] = *(const v4f*)&gate_w[i];
    }
    __syncthreads();

    const int wave = tid >> 5;          // 8 waves per block
    const int lane = tid & 31;
    const int m    = lane & 15;         // row within tile / N column for B
    const int kh   = lane >> 4;         // K-half select (0 or 1)
    const int rowBase = (blockIdx.x * 8 + wave) * 16;

    const float* __restrict__ xrow = x + (size_t)(rowBase + m) * DIM;
    // B operand column select: zero outside N<2 (W only has 2 columns).
    const float bsel = (m < 2) ? 1.0f : 0.0f;
    const int   wn   = (m < 2) ? m : 0;

    v8f acc0 = {}, acc1 = {}, acc2 = {}, acc3 = {};

    for (int k0 = 0; k0 < DIM; k0 += 16) {
        #pragma unroll
        for (int u = 0; u < 4; ++u) {
            const int k = k0 + 4 * u + 2 * kh;
            v2f a = *(const v2f*)(xrow + k);
            v2f b;
            b.x = w_lds[(k + 0) * 2 + wn] * bsel;
            b.y = w_lds[(k + 1) * 2 + wn] * bsel;
            switch (u) {
            case 0: acc0 = __builtin_amdgcn_wmma_f32_16x16x4_f32(
                        false, a, false, b, (short)0, acc0, false, false); break;
            case 1: acc1 = __builtin_amdgcn_wmma_f32_16x16x4_f32(
                        false, a, false, b, (short)0, acc1, false, false); break;
            case 2: acc2 = __builtin_amdgcn_wmma_f32_16x16x4_f32(
                        false, a, false, b, (short)0, acc2, false, false); break;
            default: acc3 = __builtin_amdgcn_wmma_f32_16x16x4_f32(
                        false, a, false, b, (short)0, acc3, false, false); break;
            }
        }
    }
    // Partial D tiles sum linearly.
    v8f acc = acc0 + acc1 + acc2 + acc3;

    // Lanes with N-column < 2 own the real logits; emit {0,1} masks.
    if (m < 2) {
        const float bb = gate_b[m];
        #pragma unroll
        for (int r = 0; r < 8; ++r) {
            const int row = rowBase + r + 8 * kh;
            const float logit = acc[r] + bb;
            // sigmoid(z) > 0.5  <=>  z > 0 (strict, matching reference)
            masks[m * NROWS + row] = (logit > 0.0f) ? 1.0f : 0.0f;
        }
    }
}

// ---------------------------------------------------------------------------
// Pass 2: streaming route/combine. float4 per thread; non-temporal stores so
// 384 MB of write-once output does not evict x (128 MB) from the 192 MB L2.
// ---------------------------------------------------------------------------
__global__ __launch_bounds__(256) void route_kernel(
    const float* __restrict__ x,
    const float* __restrict__ masks,    // [2][NROWS]
    float* __restrict__ out)            // [3][NROWS][DIM]
{
    const size_t ND = (size_t)NROWS * DIM;
    const size_t e  = ((size_t)blockIdx.x * blockDim.x + threadIdx.x) * 4;
    const int row   = (int)(e >> 12);   // DIM == 4096 == 1<<12

    const float m0 = masks[row];
    const float m1 = masks[NROWS + row];

    const v4f xv = *(const v4f*)(x + e);
    const v4f o0 = xv * m0;
    const v4f o1 = xv * m1;
    const v4f oc = xv * (m0 + m1);

    __builtin_nontemporal_store(o0, (v4f*)(out + e));
    __builtin_nontemporal_store(o1, (v4f*)(out + ND + e));
    __builtin_nontemporal_store(oc, (v4f*)(out + 2 * ND + e));
}

extern "C" void kernel_launch(void* const* d_in, const int* in_sizes, int n_in,
                              void* d_out, int out_size, void* d_ws, size_t ws_size,
                              hipStream_t stream) {
    const float* x  = (const float*)d_in[0];
    const float* gw = (const float*)d_in[1];
    const float* gb = (const float*)d_in[2];
    float* out   = (float*)d_out;
    float* masks = (float*)d_ws;        // 2 * NROWS floats = 64 KB

    // Pass 1: 8192 rows / (8 waves * 16 rows per wave) = 64 blocks.
    gate_wmma_kernel<<<NROWS / 128, 256, 0, stream>>>(x, gw, gb, masks);

    // Pass 2: NROWS*DIM/4 float4 elements / 256 threads = 32768 blocks.
    route_kernel<<<(int)(((size_t)NROWS * DIM / 4) / 256), 256, 0, stream>>>(
        x, masks, out);
}